// IntentAwareSelfAttention_20409684590919
// MI455X (gfx1250) — compile-verified
//
#include <hip/hip_runtime.h>

typedef __attribute__((ext_vector_type(2))) float v2f;
typedef __attribute__((ext_vector_type(8))) float v8f;

#define BD 4
#define SD 2048
#define ED 256
#define HD 4
#define DD 64

// D = A(16x4, f32) * B(4x16, f32) + C(16x16, f32) -- full fp32 tensor op
__device__ __forceinline__ v8f wmma4(v2f a, v2f b, v8f c) {
  return __builtin_amdgcn_wmma_f32_16x16x4_f32(
      /*neg_a=*/false, a, /*neg_b=*/false, b,
      /*c_mod=*/(short)0, c, /*reuse_a=*/false, /*reuse_b=*/false);
}

// ---------------------------------------------------------------------------
// Kernel 1: fused Q/K/V projection. One wave computes one 16x16 output tile.
// Q,K written [B][H][S][D]; V written transposed [B][H][D][S] so that the
// attention kernel's B-fragments are contiguous b64 loads.
// ---------------------------------------------------------------------------
__global__ __launch_bounds__(256) void qkv_proj_kernel(
    const float* __restrict__ his,
    const float* __restrict__ Wq, const float* __restrict__ bq,
    const float* __restrict__ Wk, const float* __restrict__ bk,
    const float* __restrict__ Wv, const float* __restrict__ bv,
    float* __restrict__ Qb, float* __restrict__ Kb, float* __restrict__ Vtb) {
  const int lane = threadIdx.x & 31;
  const int warp = threadIdx.x >> 5;
  const int lm = lane & 15;  // column / M-row selector
  const int lh = lane >> 4;  // half-wave
  const int wid = blockIdx.x * 8 + warp;
  const int mat = wid >> 13;           // 0=Q 1=K 2=V  (8192 tiles per matrix)
  const int rem = wid & 8191;
  const int r0 = (rem >> 4) << 4;      // row tile base (0..8176)
  const int c0 = (rem & 15) << 4;      // col tile base (0..240)

  const float* W = (mat == 0) ? Wq : (mat == 1) ? Wk : Wv;
  const float* bias = (mat == 0) ? bq : (mat == 1) ? bk : bv;

  v8f acc = {};
  const float* arow = his + (r0 + lm) * ED + 2 * lh;
  const float* bcol = W + (2 * lh) * ED + c0 + lm;
#pragma unroll 8
  for (int k0 = 0; k0 < ED; k0 += 4) {
    v2f a = *(const v2f*)(arow + k0);        // A[lm][k0+2lh .. +1]
    v2f b;
    b.x = bcol[k0 * ED];                      // B[k0+2lh][c0+lm]
    b.y = bcol[k0 * ED + ED];                 // B[k0+2lh+1][c0+lm]
    acc = wmma4(a, b, acc);
  }

  const float bval = bias[c0 + lm];
  const int bb = r0 >> 11;          // batch
  const int sb = r0 & (SD - 1);     // seq base
  const int h = c0 >> 6;            // head
  const int dc = c0 & (DD - 1);     // d base within head
  if (mat == 2) {
#pragma unroll
    for (int i = 0; i < 8; ++i) {
      const int m = i + 8 * lh;
      Vtb[((bb * HD + h) * DD + dc + lm) * SD + sb + m] = acc[i] + bval;
    }
  } else {
    float* dst = (mat == 0) ? Qb : Kb;
#pragma unroll
    for (int i = 0; i < 8; ++i) {
      const int m = i + 8 * lh;
      dst[((bb * HD + h) * SD + sb + m) * DD + dc + lm] = acc[i] + bval;
    }
  }
}

// ---------------------------------------------------------------------------
// Kernel 2: flash-style attention. One wave owns a 16-query tile of one
// (b,h). Each iteration consumes 64 keys (4 WMMA tiles): 64 WMMAs for QK^T,
// ONE online-softmax update (reductions amortized 4x), 64 WMMAs for P*V.
// The (1/sqrt(D))/temperature row scale is pre-folded into the Q fragments
// (A-layout holds row lane%16 in both half-waves, so the scale is per-lane).
// P is re-laid-out C->A through per-wave LDS tiles (same-wave DS ops are
// in-order; stride 18 keeps b64 alignment and spreads banks).
// ---------------------------------------------------------------------------
__global__ __launch_bounds__(256) void attn_kernel(
    const float* __restrict__ Qb, const float* __restrict__ Kb,
    const float* __restrict__ Vtb, const int* __restrict__ maskp,
    const float* __restrict__ explore, const float* __restrict__ exploit,
    float* __restrict__ attn) {
  __shared__ float pshare[8 * 4 * 16 * 18];  // 8 waves x 4 P-tiles x 16x18
  const int lane = threadIdx.x & 31;
  const int warp = threadIdx.x >> 5;
  const int lm = lane & 15;
  const int lh = lane >> 4;
  const int wid = blockIdx.x * 8 + warp;
  const int bh = wid >> 7;       // (b*H + h), 16 total
  const int qt = wid & 127;      // query tile
  const int b = bh >> 2;
  const int h = bh & 3;
  const int q0 = qt << 4;

  const float* Qp = Qb + (size_t)bh * SD * DD;
  const float* Kp = Kb + (size_t)bh * SD * DD;
  const float* Vp = Vtb + (size_t)bh * DD * SD;
  float* pt = pshare + warp * (4 * 16 * 18);

  // Per-query-row scale = (1/sqrt(D)) / temperature, folded into Q.
  // A-layout row for this lane is lm (same in both half-waves).
  float qscale;
  {
    const int mi = b * SD + q0 + lm;
    float t = 1.0f + 0.5f * explore[mi] - 0.5f * exploit[mi];
    t = fminf(fmaxf(t, 0.5f), 2.0f);
    if (maskp[mi] != 0) t = 1.0f;
    qscale = 0.125f / t;   // 1/sqrt(64) = 0.125
  }

  // Q fragments for the whole tile (16 k-steps of 4) stay in registers.
  v2f qa[16];
#pragma unroll
  for (int t = 0; t < 16; ++t) {
    v2f q = *(const v2f*)(Qp + (q0 + lm) * DD + 4 * t + 2 * lh);
    q.x *= qscale;
    q.y *= qscale;
    qa[t] = q;
  }

  float mrow[8], lrow[8];
#pragma unroll
  for (int i = 0; i < 8; ++i) { mrow[i] = -1e30f; lrow[i] = 0.0f; }
  v8f oacc[4] = {};

  const int mbase = b * SD;
  for (int kt = 0; kt < SD / 64; ++kt) {
    const int kb64 = kt << 6;

    // S = Q*K^T for 4 consecutive 16x16 key tiles (scores pre-scaled).
    v8f sc0 = {}, sc1 = {}, sc2 = {}, sc3 = {};
#pragma unroll
    for (int t = 0; t < 16; ++t) {
      const float* kr = Kp + (kb64 + lm) * DD + 4 * t + 2 * lh;
      v2f k0f = *(const v2f*)(kr);
      v2f k1f = *(const v2f*)(kr + 16 * DD);
      v2f k2f = *(const v2f*)(kr + 32 * DD);
      v2f k3f = *(const v2f*)(kr + 48 * DD);
      sc0 = wmma4(qa[t], k0f, sc0);
      sc1 = wmma4(qa[t], k1f, sc1);
      sc2 = wmma4(qa[t], k2f, sc2);
      sc3 = wmma4(qa[t], k3f, sc3);
    }

    // Key-padding mask for this lane's column in each tile.
    const int kmi = mbase + kb64 + lm;
    const bool km0 = maskp[kmi] != 0;
    const bool km1 = maskp[kmi + 16] != 0;
    const bool km2 = maskp[kmi + 32] != 0;
    const bool km3 = maskp[kmi + 48] != 0;

    // One online-softmax update covering all 64 keys.
    float alpha[8];
    float p0[8], p1[8], p2[8], p3[8];
#pragma unroll
    for (int i = 0; i < 8; ++i) {
      const float s0 = sc0[i], s1 = sc1[i], s2 = sc2[i], s3 = sc3[i];
      float sm = fmaxf(fmaxf(km0 ? -1e30f : s0, km1 ? -1e30f : s1),
                       fmaxf(km2 ? -1e30f : s2, km3 ? -1e30f : s3));
#pragma unroll
      for (int off = 1; off < 16; off <<= 1)
        sm = fmaxf(sm, __shfl_xor(sm, off, 16));
      const float mn = fmaxf(mrow[i], sm);
      const float a = __expf(mrow[i] - mn);           // 1.0 when both sentinel
      const float e0 = km0 ? 0.0f : __expf(s0 - mn);  // masked keys -> 0
      const float e1 = km1 ? 0.0f : __expf(s1 - mn);
      const float e2 = km2 ? 0.0f : __expf(s2 - mn);
      const float e3 = km3 ? 0.0f : __expf(s3 - mn);
      float rs = (e0 + e1) + (e2 + e3);
#pragma unroll
      for (int off = 1; off < 16; off <<= 1)
        rs += __shfl_xor(rs, off, 16);
      lrow[i] = lrow[i] * a + rs;
      mrow[i] = mn;
      alpha[i] = a;
      p0[i] = e0; p1[i] = e1; p2[i] = e2; p3[i] = e3;
    }

#pragma unroll
    for (int c = 0; c < 4; ++c)
#pragma unroll
      for (int i = 0; i < 8; ++i) oacc[c][i] *= alpha[i];

    // Stage 4 P tiles (C-layout) into LDS, read back as 16 A-fragments.
#pragma unroll
    for (int i = 0; i < 8; ++i) {
      const int ro = (i + 8 * lh) * 18 + lm;
      pt[ro]       = p0[i];
      pt[288 + ro] = p1[i];
      pt[576 + ro] = p2[i];
      pt[864 + ro] = p3[i];
    }
    v2f pa[16];
#pragma unroll
    for (int kk = 0; kk < 16; ++kk)
      pa[kk] = *(const v2f*)(pt + (kk >> 2) * 288 + lm * 18 + (kk & 3) * 4 +
                             2 * lh);

    // O += P * V over all 64 keys (V transposed: rows are d, cols are keys).
#pragma unroll
    for (int c = 0; c < 4; ++c) {
      v8f o = oacc[c];
      const float* vr = Vp + (c * 16 + lm) * SD + kb64 + 2 * lh;
#pragma unroll
      for (int kk = 0; kk < 16; ++kk) {
        v2f vf = *(const v2f*)(vr + 4 * kk);
        o = wmma4(pa[kk], vf, o);
      }
      oacc[c] = o;
    }
  }

  float inv[8];
#pragma unroll
  for (int i = 0; i < 8; ++i) inv[i] = 1.0f / fmaxf(lrow[i], 1e-8f);
#pragma unroll
  for (int c = 0; c < 4; ++c)
#pragma unroll
    for (int i = 0; i < 8; ++i) {
      const int sq = q0 + i + 8 * lh;
      attn[((size_t)b * SD + sq) * ED + h * DD + c * 16 + lm] =
          oacc[c][i] * inv[i];
    }
}

// ---------------------------------------------------------------------------
// Kernel 3: output projection + zero masked query rows.
// ---------------------------------------------------------------------------
__global__ __launch_bounds__(256) void out_proj_kernel(
    const float* __restrict__ attn, const float* __restrict__ Wo,
    const float* __restrict__ bo, const int* __restrict__ maskp,
    float* __restrict__ out) {
  const int lane = threadIdx.x & 31;
  const int warp = threadIdx.x >> 5;
  const int lm = lane & 15;
  const int lh = lane >> 4;
  const int wid = blockIdx.x * 8 + warp;
  const int r0 = (wid >> 4) << 4;
  const int c0 = (wid & 15) << 4;

  v8f acc = {};
  const float* arow = attn + (r0 + lm) * ED + 2 * lh;
  const float* bcol = Wo + (2 * lh) * ED + c0 + lm;
#pragma unroll 8
  for (int k0 = 0; k0 < ED; k0 += 4) {
    v2f a = *(const v2f*)(arow + k0);
    v2f b;
    b.x = bcol[k0 * ED];
    b.y = bcol[k0 * ED + ED];
    acc = wmma4(a, b, acc);
  }
  const float bval = bo[c0 + lm];
#pragma unroll
  for (int i = 0; i < 8; ++i) {
    const int row = r0 + i + 8 * lh;           // row == b*S + s
    const bool mq = maskp[row] != 0;
    out[(size_t)row * ED + c0 + lm] = mq ? 0.0f : (acc[i] + bval);
  }
}

extern "C" void kernel_launch(void* const* d_in, const int* in_sizes, int n_in,
                              void* d_out, int out_size, void* d_ws,
                              size_t ws_size, hipStream_t stream) {
  (void)in_sizes; (void)n_in; (void)out_size; (void)ws_size;
  const float* his     = (const float*)d_in[0];
  const int*   mask    = (const int*)d_in[1];
  const float* explore = (const float*)d_in[2];
  const float* exploit = (const float*)d_in[3];
  const float* Wq = (const float*)d_in[4];
  const float* bq = (const float*)d_in[5];
  const float* Wk = (const float*)d_in[6];
  const float* bk = (const float*)d_in[7];
  const float* Wv = (const float*)d_in[8];
  const float* bv = (const float*)d_in[9];
  const float* Wo = (const float*)d_in[10];
  const float* bo = (const float*)d_in[11];
  float* out = (float*)d_out;

  const size_t N = (size_t)BD * HD * SD * DD;  // 2M floats per buffer
  float* Qb   = (float*)d_ws;
  float* Kb   = Qb + N;
  float* Vtb  = Kb + N;
  float* attn = Vtb + N;

  // 3 matrices * 512 row-tiles * 16 col-tiles = 24576 wave-tiles, 8 waves/block
  qkv_proj_kernel<<<3072, 256, 0, stream>>>(his, Wq, bq, Wk, bk, Wv, bv,
                                            Qb, Kb, Vtb);
  // B*H*(S/16) = 2048 wave-tiles
  attn_kernel<<<256, 256, 0, stream>>>(Qb, Kb, Vtb, mask, explore, exploit,
                                       attn);
  // 512*16 = 8192 wave-tiles
  out_proj_kernel<<<1024, 256, 0, stream>>>(attn, Wo, bo, mask, out);
}